// ConvCaps_26706106647239
// MI455X (gfx1250) — compile-verified
//
#include <hip/hip_runtime.h>

// ---- problem constants (from reference) ------------------------------------
constexpr int BATCH = 8, BC = 32, CC = 32, KK = 3, WIN = 14, WO = 6;
constexpr int BKKN = BC * KK * KK;   // 288
constexpr int CWWN = CC * WO * WO;   // 1152
constexpr int POSN = WO * WO;        // 36
#define LOG_2PI 1.8378770664093453f

typedef float v2f __attribute__((ext_vector_type(2)));
typedef float v4f __attribute__((ext_vector_type(4)));
typedef float v8f __attribute__((ext_vector_type(8)));

// Faithful to the reference's scrambled reshape:
// patch[b,Bc,kx,ky,u,v,j,k] with flat = (u*6+v)*16 + j*4 + k
//   -> m = flat/36, wi = (flat%36)/6, wj = flat%6
//   value = pose[b,Bc,2wi+kx,2wj+ky,m], poses interpreted as (b,16,B,win,win).
__device__ __forceinline__ float pose_gather(const float* __restrict__ poses,
                                             int bb, int Bc, int kx, int ky,
                                             int flat) {
  int m = flat / POSN;
  int rem = flat - m * POSN;
  int wi = rem / WO, wj = rem - wi * WO;
  return poses[(((bb * 16 + m) * BC + Bc) * WIN + (2 * wi + kx)) * WIN +
               (2 * wj + ky)];
}

// ---- Kernel 1: votes via V_WMMA_F32_16X16X4_F32 ----------------------------
// Per bkk: D[(c,i),(bb,u,v,k)] = sum_j W[bkk,c,i,j] * patch[bkk,(u,v),j,k]
// M = 128 (8 row tiles), N = 1152 (72 col tiles), K = 4.
__global__ void __launch_bounds__(128)
votes_wmma_kernel(const float* __restrict__ poses,
                  const float* __restrict__ Wf,
                  float* __restrict__ votes) {
  const int bkk = blockIdx.x;                 // 0..287
  const int Bc = bkk / 9;
  const int kx = (bkk % 9) / 3;
  const int ky = bkk % 3;
  const int lane = threadIdx.x & 31;
  const int wave = threadIdx.x >> 5;          // 0..3
  const int mlan = lane & 15;
  const int hi   = lane >> 4;                 // lane half

  const float* Wb = Wf + bkk * (CC * 16);     // [c][i][j]

  // A tiles: 16x4 f32, lane m = mlan, K = vgpr + 2*hi (ISA 7.12.2)
  v2f a[8];
#pragma unroll
  for (int rt = 0; rt < 8; ++rt) {
    int row = rt * 16 + mlan;                 // (c,i)
    int c = row >> 2, i = row & 3;
    int k = hi * 2;
    a[rt].x = Wb[c * 16 + i * 4 + k];
    a[rt].y = Wb[c * 16 + i * 4 + k + 1];
  }

  for (int ct = wave; ct < 72; ct += 4) {     // uniform per wave (EXEC all 1s)
    int col = ct * 16 + mlan;                 // N index: ((bb*6+u)*6+v)*4 + k
    int kc  = col & 3;
    int pos = col >> 2;
    int bb  = pos / POSN;
    int pout = pos - bb * POSN;               // u*6 + v

    // B tile 4x16: rows K=j, K = vgpr + 2*hi
    int j0 = hi * 2;
    v2f bm;
    bm.x = pose_gather(poses, bb, Bc, kx, ky, pout * 16 + j0 * 4 + kc);
    bm.y = pose_gather(poses, bb, Bc, kx, ky, pout * 16 + (j0 + 1) * 4 + kc);

#pragma unroll
    for (int rt = 0; rt < 8; ++rt) {
      v8f cz = {};
      v8f d = __builtin_amdgcn_wmma_f32_16x16x4_f32(
          false, a[rt], false, bm, (short)0, cz, false, false);
#pragma unroll
      for (int v = 0; v < 8; ++v) {           // D: M = v + 8*hi, N = mlan
        int row = rt * 16 + v + hi * 8;
        int c = row >> 2, i = row & 3;
        votes[((bb * BKKN + bkk) * CWWN + (c * POSN + pout)) * 16 + i * 4 + kc] =
            d[v];
      }
    }
  }
}

// ---- Kernel 2: activation patches (NOT scrambled) --------------------------
__global__ void __launch_bounds__(256)
actp_kernel(const float* __restrict__ acts, float* __restrict__ actp) {
  int idx = blockIdx.x * blockDim.x + threadIdx.x;   // b*BKKN*POSN exact
  int pos = idx % POSN;
  int t = idx / POSN;
  int bkk = t % BKKN;
  int b = t / BKKN;
  int Bc = bkk / 9, kx = (bkk % 9) / 3, ky = bkk % 3;
  int u = pos / WO, v = pos % WO;
  actp[idx] = acts[((b * BC + Bc) * WIN + (2 * u + kx)) * WIN + (2 * v + ky)];
}

// ---- Kernel 3: per-(b,cww) EM stats: sum_R, mu, sigma^2 --------------------
// Hottest kernel: streams the full votes tensor; votes reads are b128.
__global__ void __launch_bounds__(128)
stats_kernel(const float* __restrict__ votes, const float* __restrict__ actp,
             const float* __restrict__ ap, const float* __restrict__ rowsum,
             float* __restrict__ mu_ws, float* __restrict__ iss_ws,
             float* __restrict__ hld_ws, const float* __restrict__ beta_v,
             const float* __restrict__ beta_a,
             const float* __restrict__ lambda_, float* __restrict__ out_mu,
             float* __restrict__ out_a, int iter, int is_final) {
  __shared__ __align__(16) float s_v[BKKN * 16];   // 18432 B
  __shared__ float s_ra[BKKN];
  __shared__ float s_red[128];
  __shared__ float s_mu[16];
  __shared__ float s_sumR;

  const int tid = threadIdx.x;
  const int b = blockIdx.x / CWWN;
  const int cww = blockIdx.x % CWWN;
  const int pos = cww % POSN;
  const int e = tid & 15;            // pose element
  const int g = tid >> 4;            // 0..7 reduction groups

  // 128-bit vectorized fill: 1152 x 16B chunks, 9 iterations/thread,
  // consecutive threads cover contiguous 64B per bkk row.
  v4f* s_v4 = (v4f*)s_v;
  for (int f = tid; f < BKKN * 4; f += 128) {
    int bkk = f >> 2, q = f & 3;
    s_v4[f] = *(const v4f*)(votes +
                            (((size_t)(b * BKKN + bkk) * CWWN + cww) << 4) +
                            (q << 2));
  }
  for (int bkk = tid; bkk < BKKN; bkk += 128) {
    float R = (iter == 0)
                  ? (1.0f / CWWN)
                  : ap[(b * BKKN + bkk) * CWWN + cww] / rowsum[b * BKKN + bkk];
    s_ra[bkk] = R * actp[(b * BKKN + bkk) * POSN + pos];
  }
  __syncthreads();

  float part = 0.f;
  for (int bkk = g; bkk < BKKN; bkk += 8) part += s_ra[bkk] * s_v[bkk * 16 + e];
  s_red[tid] = part;
  if (tid == 0) {
    float s = 0.f;
    for (int k = 0; k < BKKN; ++k) s += s_ra[k];
    s_sumR = s;
  }
  __syncthreads();
  const float sumR = s_sumR;
  if (tid < 16) {
    float m = 0.f;
    for (int gg = 0; gg < 8; ++gg) m += s_red[gg * 16 + tid];
    s_mu[tid] = m / sumR;
  }
  __syncthreads();
  const float mu_e = s_mu[e];
  part = 0.f;
  for (int bkk = g; bkk < BKKN; bkk += 8) {
    float d = s_v[bkk * 16 + e] - mu_e;
    part += s_ra[bkk] * d * d;
  }
  s_red[tid] = part;
  __syncthreads();
  if (tid < 16) {
    float ssq = 0.f;
    for (int gg = 0; gg < 8; ++gg) ssq += s_red[gg * 16 + tid];
    ssq /= sumR;
    if (is_final) {
      out_mu[(b * CWWN + cww) * 16 + tid] = s_mu[tid];
    } else {
      mu_ws[(b * CWWN + cww) * 16 + tid] = s_mu[tid];
      iss_ws[(b * CWWN + cww) * 16 + tid] = 1.0f / ssq;
    }
    s_red[tid] = logf(ssq);   // safe: thread e only touches column e
  }
  __syncthreads();
  if (tid == 0) {
    float ls = 0.f;
    for (int k = 0; k < 16; ++k) ls += s_red[k];
    if (is_final) {
      float cost = sumR * (16.0f * beta_v[cww] + ls);
      float x = lambda_[0] * (beta_a[cww] - cost);
      out_a[b * CWWN + cww] = 1.0f / (1.0f + expf(-x));
    } else {
      hld_ws[b * CWWN + cww] = 0.5f * ls;    // 0.5 * sum_k log(sigma^2_k)
    }
  }
}

// ---- Kernel 4: R numerator ap = act * exp(sum_k log_p_k) -------------------
__global__ void __launch_bounds__(256)
ap_kernel(const float* __restrict__ votes, const float* __restrict__ actp,
          const float* __restrict__ mu_ws, const float* __restrict__ iss_ws,
          const float* __restrict__ hld_ws, float* __restrict__ ap) {
  int idx = blockIdx.x * blockDim.x + threadIdx.x;  // (b,bkk,cww), cww fastest
  int cww = idx % CWWN;
  int t = idx / CWWN;
  int bkk = t % BKKN;
  int b = t / BKKN;
  const v4f* v4  = (const v4f*)(votes + (size_t)idx * 16);   // 64B contiguous
  const v4f* mu4 = (const v4f*)(mu_ws + (size_t)(b * CWWN + cww) * 16);
  const v4f* is4 = (const v4f*)(iss_ws + (size_t)(b * CWWN + cww) * 16);
  v4f acc = {};
#pragma unroll
  for (int q = 0; q < 4; ++q) {
    v4f d = v4[q] - mu4[q];
    acc += d * d * is4[q];
  }
  float s = acc.x + acc.y + acc.z + acc.w;
  float lp = -0.5f * s - hld_ws[b * CWWN + cww] - 8.0f * LOG_2PI;
  ap[idx] = actp[(b * BKKN + bkk) * POSN + (cww % POSN)] * expf(lp);
}

// ---- Kernel 5: row sums of ap over Cww -------------------------------------
__global__ void __launch_bounds__(256)
rowsum_kernel(const float* __restrict__ ap, float* __restrict__ rowsum) {
  __shared__ float sred[256];
  int row = blockIdx.x;                              // b*BKKN + bkk
  const float* r = ap + (size_t)row * CWWN;
  float s = 0.f;
  for (int c = threadIdx.x; c < CWWN; c += 256) s += r[c];
  sred[threadIdx.x] = s;
  __syncthreads();
  for (int off = 128; off > 0; off >>= 1) {
    if (threadIdx.x < off) sred[threadIdx.x] += sred[threadIdx.x + off];
    __syncthreads();
  }
  if (threadIdx.x == 0) rowsum[row] = sred[0];
}

// ---- host-side orchestration ------------------------------------------------
extern "C" void kernel_launch(void* const* d_in, const int* in_sizes, int n_in,
                              void* d_out, int out_size, void* d_ws,
                              size_t ws_size, hipStream_t stream) {
  const float* poses   = (const float*)d_in[0];
  const float* acts    = (const float*)d_in[1];
  const float* lambda_ = (const float*)d_in[2];
  const float* Wf      = (const float*)d_in[3];
  const float* beta_v  = (const float*)d_in[4];
  const float* beta_a  = (const float*)d_in[5];

  const size_t N_VOTES = (size_t)BATCH * BKKN * CWWN * 16;  // 42,467,328
  const size_t N_ACTP  = (size_t)BATCH * BKKN * POSN;       // 82,944
  const size_t N_AP    = (size_t)BATCH * BKKN * CWWN;       // 2,654,208
  const size_t N_ROWS  = (size_t)BATCH * BKKN;              // 2,304
  const size_t N_MU    = (size_t)BATCH * CWWN * 16;         // 147,456

  float* ws     = (float*)d_ws;                             // ~174 MiB total
  float* votes  = ws;
  float* actp   = votes + N_VOTES;
  float* apbuf  = actp + N_ACTP;
  float* rowsum = apbuf + N_AP;
  float* mu_ws  = rowsum + N_ROWS;
  float* iss_ws = mu_ws + N_MU;
  float* hld_ws = iss_ws + N_MU;

  float* out_mu = (float*)d_out;                 // (b, Cww, 16) == (b,C,w,w,16)
  float* out_a  = out_mu + N_MU;                 // (b, Cww)     == (b,C,w,w)

  votes_wmma_kernel<<<BKKN, 128, 0, stream>>>(poses, Wf, votes);
  actp_kernel<<<(int)(N_ACTP / 256), 256, 0, stream>>>(acts, actp);

  for (int it = 0; it < 3; ++it) {
    stats_kernel<<<BATCH * CWWN, 128, 0, stream>>>(
        votes, actp, apbuf, rowsum, mu_ws, iss_ws, hld_ws, beta_v, beta_a,
        lambda_, out_mu, out_a, it, 0);
    ap_kernel<<<(int)(N_AP / 256), 256, 0, stream>>>(votes, actp, mu_ws,
                                                     iss_ws, hld_ws, apbuf);
    rowsum_kernel<<<(int)N_ROWS, 256, 0, stream>>>(apbuf, rowsum);
  }
  stats_kernel<<<BATCH * CWWN, 128, 0, stream>>>(
      votes, actp, apbuf, rowsum, mu_ws, iss_ws, hld_ws, beta_v, beta_a,
      lambda_, out_mu, out_a, 3, 1);
}